// DistributionEncoder_29781303230995
// MI455X (gfx1250) — compile-verified
//
#include <hip/hip_runtime.h>
#include <hip/hip_bf16.h>

// ---------------------------------------------------------------------------
// DistributionEncoder for MI455X (gfx1250, wave32, WMMA)
//   x:[64,65536] int32 bytes -> feats[64,321] -> 3-layer MLP -> out[64,256] f32
// ---------------------------------------------------------------------------

#define B_ROWS   64
#define L_LEN    65536
#define WIN      64
#define STRIDE   32
#define NW       2047          // (65536-64)/32 + 1
#define CDFP     64
#define KPAD     324           // 321 padded to multiple of 4 for WMMA k-steps
#define HID      512
#define DMODEL   256

typedef float v2f __attribute__((ext_vector_type(2)));
typedef float v8f __attribute__((ext_vector_type(8)));

// ---------------------------------------------------------------------------
// K1: window entropies. Block = 256 threads = 256 windows of one row.
//     ent(w) = -(1/64) * sum_i log2(c_i/64 + 1e-5), c_i = count of byte at pos i.
//     (total + 1e-10 == 64 exactly in f32, so 1/64 is exact.)
// ---------------------------------------------------------------------------
__global__ void window_entropy_kernel(const int* __restrict__ x,
                                      float* __restrict__ went)
{
    __shared__ unsigned wbytes[2056];         // 255*8 + 16 words of packed bytes
    const int row = blockIdx.y;
    const int w0  = blockIdx.x * 256;
    const int t   = threadIdx.x;
    const int s0  = w0 * STRIDE;              // first byte this block needs
    const int* xr = x + row * L_LEN;

    // Pack int32 byte values into LDS words (4 bytes per word).
    for (int wd = t; wd < 2056; wd += 256) {
        int e = s0 + wd * 4;
        unsigned p = 0u;
        #pragma unroll
        for (int b = 0; b < 4; ++b) {
            int idx = e + b;
            if (idx < L_LEN) {
                int v = xr[idx];
                v = v < 0 ? 0 : (v > 255 ? 255 : v);
                p |= ((unsigned)v) << (8 * b);
            }
        }
        wbytes[wd] = p;
    }
    __syncthreads();

    const int w = w0 + t;
    if (w < NW) {
        unsigned wv[16];
        const int base = t * 8;               // window byte offset t*32 -> word t*8
        #pragma unroll
        for (int j = 0; j < 16; ++j) wv[j] = wbytes[base + j];

        float s = 0.0f;
        for (int j = 0; j < 16; ++j) {        // word holding the 4 query bytes
            unsigned pw = wbytes[base + j];   // LDS read (dynamic idx ok)
            #pragma unroll
            for (int by = 0; by < 4; ++by) {
                unsigned m = ((pw >> (8 * by)) & 0xFFu) * 0x01010101u;
                int c = 0;
                #pragma unroll
                for (int jj = 0; jj < 16; ++jj) {   // count matches in 64 bytes
                    unsigned d = wv[jj] ^ m;
                    unsigned z = (d - 0x01010101u) & (~d) & 0x80808080u;
                    c += __popc(z);
                }
                s += log2f((float)c * 0.015625f + 1e-5f);
            }
        }
        went[row * NW + w] = -0.015625f * s;
    }
}

// ---------------------------------------------------------------------------
// K2: per-row global histogram (8 wave-private LDS sub-hists), normalized
//     histogram -> feats[:,0:256], entropy -> feats[:,256].
// ---------------------------------------------------------------------------
__global__ void hist_kernel(const int* __restrict__ x,
                            float* __restrict__ feats)
{
    __shared__ unsigned sub[8 * 256];
    __shared__ float    red[8];
    const int row  = blockIdx.x;
    const int t    = threadIdx.x;
    const int wave = t >> 5;
    const int* xr  = x + row * L_LEN;

    for (int i = t; i < 8 * 256; i += 256) sub[i] = 0u;
    __syncthreads();

    for (int i = t; i < L_LEN; i += 256) {
        int v = xr[i];
        v = v < 0 ? 0 : (v > 255 ? 255 : v);
        atomicAdd(&sub[wave * 256 + v], 1u);
    }
    __syncthreads();

    unsigned c = 0;
    #pragma unroll
    for (int wv = 0; wv < 8; ++wv) c += sub[wv * 256 + t];

    const float p = (float)c * (1.0f / 65536.0f);  // 65536 + 1e-10 == 65536 in f32
    feats[row * KPAD + t] = p;

    float term = p * log2f(p + 1e-5f);
    #pragma unroll
    for (int o = 16; o > 0; o >>= 1) term += __shfl_xor(term, o, 32);
    if ((t & 31) == 0) red[wave] = term;
    __syncthreads();
    if (t == 0) {
        float s = 0.0f;
        #pragma unroll
        for (int wv = 0; wv < 8; ++wv) s += red[wv];
        feats[row * KPAD + 256] = -s;
    }
}

// ---------------------------------------------------------------------------
// K3: empirical CDF at levels j*8/63 -> feats[:,257:321]; zero-pad cols 321..323.
// ---------------------------------------------------------------------------
__global__ void cdf_kernel(const float* __restrict__ went,
                           float* __restrict__ feats)
{
    __shared__ float buf[256];
    const int row = blockIdx.x;
    const int t   = threadIdx.x;
    const float* wr = went + row * NW;

    const float level = (float)(t & 63) * (8.0f / 63.0f);
    int cnt = 0;
    for (int base = 0; base < NW; base += 256) {
        int i = base + t;
        buf[t] = (i < NW) ? wr[i] : 0.0f;
        __syncthreads();
        const int n = (NW - base) < 256 ? (NW - base) : 256;
        if (t < CDFP) {
            for (int j = 0; j < n; ++j) cnt += (buf[j] <= level) ? 1 : 0;
        }
        __syncthreads();
    }
    if (t < CDFP)             feats[row * KPAD + 257 + t] = (float)cnt * (1.0f / 2047.0f);
    if (t >= 64 && t < 67)    feats[row * KPAD + 321 + (t - 64)] = 0.0f;  // K padding
}

// ---------------------------------------------------------------------------
// Weight transpose + zero-pad: Wt[n*KP + k] = (k < Kvalid) ? W[k*N + n] : 0.
// Makes the WMMA B fragment contiguous along k (one b64 load per lane) and
// removes all K-guards from the GEMM hot loop. Writes are fully coalesced;
// strided reads hit L2 (weights are <= 1 MB, L2 is 192 MB).
// ---------------------------------------------------------------------------
__global__ void transpose_pad_kernel(const float* __restrict__ W,
                                     float* __restrict__ Wt,
                                     int N, int Kvalid, int KP)
{
    const int idx = blockIdx.x * 256 + threadIdx.x;
    if (idx >= N * KP) return;
    const int n = idx / KP;
    const int k = idx - n * KP;
    Wt[idx] = (k < Kvalid) ? W[k * N + n] : 0.0f;
}

// ---------------------------------------------------------------------------
// GEMM via V_WMMA_F32_16X16X4_F32. Each wave owns one or more 16x16 tiles
// (tile index uniform per wave -> EXEC stays all-ones around every WMMA).
//   A : [64, lda] row-major, valid/zero-padded up to K
//   Bt: [N, K]    transposed weights (k contiguous), zero-padded up to K
//   C : [64, ldc] = A*B + bias
// Fragment layouts per CDNA5 ISA 7.12.2 (f32 16x16x4):
//   A VGPR r: lanes0-15 -> A[m][k0+r], lanes16-31 -> A[m][k0+2+r]
//   B VGPR r: lanes0-15 -> B[k0+r][n], lanes16-31 -> B[k0+2+r][n]
//   D VGPR r: lanes0-15 -> D[m0+r][n], lanes16-31 -> D[m0+8+r][n]
// ---------------------------------------------------------------------------
__global__ void gemm_wmma_kernel(const float* __restrict__ A, int lda,
                                 const float* __restrict__ Bt,
                                 const float* __restrict__ bias,
                                 float* __restrict__ C, int ldc,
                                 int N, int K)
{
    const int lane = threadIdx.x & 31;
    const int wave = threadIdx.x >> 5;
    const int hl   = lane >> 4;                    // half-wave select
    const int l16  = lane & 15;
    const int nTilesN  = N >> 4;
    const int numTiles = (B_ROWS >> 4) * nTilesN;
    const int tile0    = blockIdx.x * 8 + wave;
    const int tStride  = gridDim.x * 8;

    for (int tile = tile0; tile < numTiles; tile += tStride) {
        const int mT  = tile / nTilesN;
        const int nT  = tile - mT * nTilesN;
        const int row = mT * 16 + l16;
        const int col = nT * 16 + l16;

        const float* __restrict__ aRow = A  + row * lda;   // 8B-aligned (lda even)
        const float* __restrict__ bCol = Bt + col * K;     // 8B-aligned (K even)

        v8f acc{};
        #pragma unroll 4
        for (int k = 0; k < K; k += 4) {
            const int ka = k + 2 * hl;
            const v2f a = *(const v2f*)(aRow + ka);        // A[m][ka..ka+1]
            const v2f b = *(const v2f*)(bCol + ka);        // B[ka..ka+1][n]
            acc = __builtin_amdgcn_wmma_f32_16x16x4_f32(
                    /*neg_a=*/false, a, /*neg_b=*/false, b,
                    /*c_mod=*/(short)0, acc, /*reuse_a=*/false, /*reuse_b=*/false);
        }

        const float bs = bias[col];
        #pragma unroll
        for (int r = 0; r < 8; ++r) {
            const int m = mT * 16 + r + 8 * hl;
            C[m * ldc + col] = acc[r] + bs;
        }
    }
}

// ---------------------------------------------------------------------------
// LayerNorm (biased var) + ReLU, in place. One block per row, 256 thr, N=512.
// ---------------------------------------------------------------------------
__global__ void ln_relu_kernel(float* __restrict__ H,
                               const float* __restrict__ g,
                               const float* __restrict__ be)
{
    __shared__ float rs[8], rq[8];
    const int row  = blockIdx.x;
    const int t    = threadIdx.x;
    const int wave = t >> 5;
    float* h = H + row * HID;

    const float v0 = h[t], v1 = h[t + 256];
    float s = v0 + v1;
    float q = v0 * v0 + v1 * v1;
    #pragma unroll
    for (int o = 16; o > 0; o >>= 1) {
        s += __shfl_xor(s, o, 32);
        q += __shfl_xor(q, o, 32);
    }
    if ((t & 31) == 0) { rs[wave] = s; rq[wave] = q; }
    __syncthreads();
    float tot = 0.0f, totq = 0.0f;
    #pragma unroll
    for (int wv = 0; wv < 8; ++wv) { tot += rs[wv]; totq += rq[wv]; }

    const float mu  = tot * (1.0f / (float)HID);
    float var = totq * (1.0f / (float)HID) - mu * mu;
    var = var < 0.0f ? 0.0f : var;
    const float inv = rsqrtf(var + 1e-5f);

    const float o0 = (v0 - mu) * inv * g[t] + be[t];
    const float o1 = (v1 - mu) * inv * g[t + 256] + be[t + 256];
    h[t]       = o0 > 0.0f ? o0 : 0.0f;
    h[t + 256] = o1 > 0.0f ? o1 : 0.0f;
}

// ---------------------------------------------------------------------------
extern "C" void kernel_launch(void* const* d_in, const int* in_sizes, int n_in,
                              void* d_out, int out_size, void* d_ws, size_t ws_size,
                              hipStream_t stream) {
    const int*   x   = (const int*)  d_in[0];
    const float* W1  = (const float*)d_in[1];
    const float* b1  = (const float*)d_in[2];
    const float* g1  = (const float*)d_in[3];
    const float* be1 = (const float*)d_in[4];
    const float* W2  = (const float*)d_in[5];
    const float* b2  = (const float*)d_in[6];
    const float* g2  = (const float*)d_in[7];
    const float* be2 = (const float*)d_in[8];
    const float* W3  = (const float*)d_in[9];
    const float* b3  = (const float*)d_in[10];
    float* out = (float*)d_out;

    // Workspace layout (floats):
    //   went[64*2048] | feats[64*324] | h1[64*512] | h2[64*512]
    //   | W1t[512*324] | W2t[512*512] | W3t[256*512]
    float* went  = (float*)d_ws;
    float* feats = went  + B_ROWS * 2048;
    float* h1    = feats + B_ROWS * KPAD;
    float* h2    = h1    + B_ROWS * HID;
    float* W1t   = h2    + B_ROWS * HID;
    float* W2t   = W1t   + HID * KPAD;
    float* W3t   = W2t   + HID * HID;

    // Feature pipeline.
    window_entropy_kernel<<<dim3(8, B_ROWS), 256, 0, stream>>>(x, went);
    hist_kernel<<<B_ROWS, 256, 0, stream>>>(x, feats);
    cdf_kernel<<<B_ROWS, 256, 0, stream>>>(went, feats);

    // Weight transposes (zero-padded along k).
    transpose_pad_kernel<<<(HID * KPAD + 255) / 256, 256, 0, stream>>>(W1, W1t, HID, 321, KPAD);
    transpose_pad_kernel<<<(HID * HID  + 255) / 256, 256, 0, stream>>>(W2, W2t, HID, HID, HID);
    transpose_pad_kernel<<<(DMODEL * HID + 255) / 256, 256, 0, stream>>>(W3, W3t, DMODEL, HID, HID);

    // feats[64,324] @ W1 (+b1) -> h1 ; LN+ReLU.   128 tiles -> 16 blocks x 8 waves.
    gemm_wmma_kernel<<<16, 256, 0, stream>>>(feats, KPAD, W1t, b1, h1, HID, HID, KPAD);
    ln_relu_kernel<<<B_ROWS, 256, 0, stream>>>(h1, g1, be1);

    // h1 @ W2 (+b2) -> h2 ; LN+ReLU.              128 tiles -> 16 blocks.
    gemm_wmma_kernel<<<16, 256, 0, stream>>>(h1, HID, W2t, b2, h2, HID, HID, HID);
    ln_relu_kernel<<<B_ROWS, 256, 0, stream>>>(h2, g2, be2);

    // h2 @ W3 (+b3) -> out.                       64 tiles -> 8 blocks.
    gemm_wmma_kernel<<<8, 256, 0, stream>>>(h2, HID, W3t, b3, out, DMODEL, DMODEL, HID);
}